// PostProcessor_60713657696861
// MI455X (gfx1250) — compile-verified
//
#include <hip/hip_runtime.h>
#include <hip/hip_bf16.h>
#include <math.h>

// ---------------------------------------------------------------------------
// FCOS-style detection post-processing for MI455X (gfx1250, wave32).
// Phases:
//   K0 zero    : zero the padded per-batch arrays (poison-safe).
//   K1 score   : streaming pass, score = sqrt(max_fg * sigmoid(cnt)).  ~73MB,
//                HBM-bound (~3us @ 23.3 TB/s) -> plain coalesced loads.
//   K2 topk    : per-batch LDS histogram threshold + 4096-wide bitonic sort.
//   K3 gather  : recompute class/box at the 1000 selected positions; bake the
//                class offset (+cls*1e5) into "offset boxes" + areas (matches
//                the reference's fp32 rounding exactly).
//   K4 ioumask : pairwise IOU > 0.6 suppression bitmask. The denominator
//                area_i + area_j outer-sum per 16x16 tile is computed with
//                V_WMMA_F32_16X16X4_F32 (rank-2 matmul A=[area,1], B=[1;area]);
//                wave32 __ballot assembles 16 suppression bits/row.
//   K5 nms     : sequential greedy sweep, 1 wave32/batch, 32-lane-resident
//                1024-bit suppression set, __shfl broadcast, prefetch next row.
// ---------------------------------------------------------------------------

#define NPOS   21824      // 128^2 + 64^2 + 32^2 + 16^2 + 8^2
#define BATCH  32
#define NCLS   21
#define KTOP   1000
#define KPAD   1024
#define CAND   4096
#define NBIN   1024
#define CLS_THR 0.05f
#define IOU_THR 0.6f
#define CLASS_OFFSET 100000.0f

typedef __attribute__((ext_vector_type(2))) float v2f;
typedef __attribute__((ext_vector_type(8))) float v8f;

struct DetPtrs {
    const float* cls[5];
    const float* cnt[5];
    const float* reg[5];
};

__device__ __forceinline__ void decode_level(int n, int& lv, int& p, int& h, int& s) {
    if (n < 16384)      { lv = 0; p = n;         h = 128; s = 8;   }
    else if (n < 20480) { lv = 1; p = n - 16384; h = 64;  s = 16;  }
    else if (n < 21504) { lv = 2; p = n - 20480; h = 32;  s = 32;  }
    else if (n < 21760) { lv = 3; p = n - 21504; h = 16;  s = 64;  }
    else                { lv = 4; p = n - 21760; h = 8;   s = 128; }
}

__device__ __forceinline__ float fg_argmax(const float* c0, size_t hs, int& bc) {
    float best = c0[hs];          // class 1 -> fg index 0
    bc = 0;
    #pragma unroll
    for (int c = 2; c < NCLS; ++c) {
        float v = c0[(size_t)c * hs];
        if (v > best) { best = v; bc = c - 1; }   // strict >: first max wins (argmax)
    }
    return best;
}

// --------------------------------------------------------------------- K0
__global__ void zero_kernel(float* topv, float* areas, float* boxes_off) {
    int i = blockIdx.x * blockDim.x + threadIdx.x;
    if (i < BATCH * KPAD) { topv[i] = 0.f; areas[i] = 0.f; }
    if (i < BATCH * KPAD * 4) boxes_off[i] = 0.f;
}

// --------------------------------------------------------------------- K1
__global__ __launch_bounds__(256) void score_kernel(DetPtrs P, float* __restrict__ scores) {
    int n = blockIdx.x * blockDim.x + threadIdx.x;
    int b = blockIdx.y;
    if (n >= NPOS) return;
    int lv, p, h, s;
    decode_level(n, lv, p, h, s);
    size_t hs = (size_t)h * h;
    const float* c0 = P.cls[lv] + (size_t)b * NCLS * hs + p;
    int bc;
    float best = fg_argmax(c0, hs, bc);
    float cl = P.cnt[lv][(size_t)b * hs + p];
    float cp = 1.f / (1.f + __expf(-cl));              // sigmoid
    scores[(size_t)b * NPOS + n] = sqrtf(best * cp);
}

// --------------------------------------------------------------------- K2
__global__ __launch_bounds__(256) void topk_kernel(const float* __restrict__ scores,
                                                   float* __restrict__ topv,
                                                   int* __restrict__ topi) {
    __shared__ int   hist[NBIN];
    __shared__ float cs[CAND];
    __shared__ int   ci[CAND];
    __shared__ int   s_nc;
    __shared__ int   s_bstar;
    int b = blockIdx.x, t = threadIdx.x;
    const float* sc = scores + (size_t)b * NPOS;

    for (int i = t; i < NBIN; i += 256) hist[i] = 0;
    if (t == 0) s_nc = 0;
    __syncthreads();

    for (int n = t; n < NPOS; n += 256) {
        int bin = (int)(sc[n] * (float)NBIN);
        bin = min(max(bin, 0), NBIN - 1);
        atomicAdd(&hist[bin], 1);
    }
    __syncthreads();

    if (t == 0) {
        int acc = 0, bstar = 0;
        for (int i = NBIN - 1; i >= 0; --i) {
            acc += hist[i];
            if (acc >= KTOP) { bstar = i; break; }
        }
        s_bstar = bstar;
    }
    __syncthreads();
    int bstar = s_bstar;

    for (int n = t; n < NPOS; n += 256) {
        float v = sc[n];
        int bin = min(max((int)(v * (float)NBIN), 0), NBIN - 1);
        if (bin >= bstar) {
            int k = atomicAdd(&s_nc, 1);
            if (k < CAND) { cs[k] = v; ci[k] = n; }
        }
    }
    __syncthreads();
    int nc = min(s_nc, CAND);
    for (int k = t; k < CAND; k += 256)
        if (k >= nc) { cs[k] = -1e30f; ci[k] = 0x7FFFFFFF; }
    __syncthreads();

    // bitonic sort, descending score, ascending index on ties (matches top_k)
    for (int kk = 2; kk <= CAND; kk <<= 1) {
        for (int j = kk >> 1; j > 0; j >>= 1) {
            for (int idx = t; idx < CAND; idx += 256) {
                int ixj = idx ^ j;
                if (ixj > idx) {
                    bool dir = ((idx & kk) == 0);
                    float s1 = cs[idx], s2 = cs[ixj];
                    int   i1 = ci[idx], i2 = ci[ixj];
                    bool before = (s2 > s1) || (s2 == s1 && i2 < i1);  // ixj ranks before idx
                    if (before == dir) {
                        cs[idx] = s2; cs[ixj] = s1;
                        ci[idx] = i2; ci[ixj] = i1;
                    }
                }
            }
            __syncthreads();
        }
    }
    for (int r = t; r < KTOP; r += 256) {
        topv[(size_t)b * KPAD + r] = cs[r];
        topi[(size_t)b * KPAD + r] = ci[r];
    }
}

// --------------------------------------------------------------------- K3
__global__ __launch_bounds__(256) void gather_kernel(DetPtrs P,
                                                     const int* __restrict__ topi,
                                                     float* __restrict__ boxes_raw,
                                                     float* __restrict__ boxes_off,
                                                     float* __restrict__ areas,
                                                     int* __restrict__ classes) {
    int k = blockIdx.x * blockDim.x + threadIdx.x;
    int b = blockIdx.y;
    if (k >= KTOP) return;
    int n = topi[(size_t)b * KPAD + k];
    int lv, p, h, s;
    decode_level(n, lv, p, h, s);
    size_t hs = (size_t)h * h;
    int row = p / h, col = p % h;
    float cx = (float)(col * s + s / 2);
    float cy = (float)(row * s + s / 2);
    const float* rg = P.reg[lv] + (size_t)b * 4 * hs + p;
    float r0 = rg[0], r1 = rg[hs], r2 = rg[2 * hs], r3 = rg[3 * hs];
    float x1 = cx - r0, y1 = cy - r1, x2 = cx + r2, y2 = cy + r3;

    const float* c0 = P.cls[lv] + (size_t)b * NCLS * hs + p;
    int bc;
    (void)fg_argmax(c0, hs, bc);
    float off = (float)bc * CLASS_OFFSET;

    size_t o = ((size_t)b * KPAD + k) * 4;
    boxes_raw[o + 0] = x1; boxes_raw[o + 1] = y1;
    boxes_raw[o + 2] = x2; boxes_raw[o + 3] = y2;
    // class offset baked in BEFORE area/IOU, mirroring the reference's fp32 math
    float ox1 = x1 + off, oy1 = y1 + off, ox2 = x2 + off, oy2 = y2 + off;
    boxes_off[o + 0] = ox1; boxes_off[o + 1] = oy1;
    boxes_off[o + 2] = ox2; boxes_off[o + 3] = oy2;
    areas[(size_t)b * KPAD + k] = (ox2 - ox1) * (oy2 - oy1);
    classes[(size_t)b * KPAD + k] = bc;
}

// --------------------------------------------------------------------- K4
// One wave32 per 16-row x 32-col tile. Denominator outer-sum via f32 WMMA.
__global__ __launch_bounds__(32) void ioumask_kernel(const float* __restrict__ boxes_off,
                                                     const float* __restrict__ areas,
                                                     unsigned* __restrict__ mask) {
    int tj = blockIdx.x;                 // 0..31  -> cols j0..j0+31
    int ti = blockIdx.y;                 // 0..62  -> rows i0..i0+15
    int b  = blockIdx.z;
    int lane = threadIdx.x;              // wave32
    int i0 = ti * 16, j0 = tj * 32;

    const float* ar = areas     + (size_t)b * KPAD;
    const float* bx = boxes_off + (size_t)b * KPAD * 4;

    int  li = lane & 15;
    bool lo = lane < 16;
    float areaI  = ar[i0 + li];
    float areaJL = ar[j0 + li];
    float areaJR = ar[j0 + 16 + li];

    // A (16x4): col0 = area_i, col1 = 1. B (4x16): row0 = 1, row1 = area_j.
    // Lanes 0..15 hold K=0,1; lanes 16..31 hold K=2,3 (zeros).
    v2f a, bl, br;
    a.x  = lo ? areaI  : 0.f;  a.y  = lo ? 1.f    : 0.f;
    bl.x = lo ? 1.f    : 0.f;  bl.y = lo ? areaJL : 0.f;
    br.x = lo ? 1.f    : 0.f;  br.y = lo ? areaJR : 0.f;
    v8f c = {0.f, 0.f, 0.f, 0.f, 0.f, 0.f, 0.f, 0.f};
    // D[M][N] = area_i[i0+M] + area_j[j0+N]; EXEC is all-1s here (no branches).
    v8f dL = __builtin_amdgcn_wmma_f32_16x16x4_f32(false, a, false, bl, (short)0, c, false, false);
    v8f dR = __builtin_amdgcn_wmma_f32_16x16x4_f32(false, a, false, br, (short)0, c, false, false);

    // Column boxes for this lane (col N = lane&15 in each 16-wide half-tile).
    const float* cbL = bx + (size_t)(j0 + li) * 4;
    const float* cbR = bx + (size_t)(j0 + 16 + li) * 4;
    float jx1L = cbL[0], jy1L = cbL[1], jx2L = cbL[2], jy2L = cbL[3];
    float jx1R = cbR[0], jy1R = cbR[1], jx2R = cbR[2], jy2R = cbR[3];

    int hi = lane >> 4;
    unsigned balL[8], balR[8];
    #pragma unroll
    for (int v = 0; v < 8; ++v) {
        int irow = i0 + v + 8 * hi;      // C/D layout: VGPR v, half-wave hi
        const float* rb = bx + (size_t)irow * 4;
        float ix1 = rb[0], iy1 = rb[1], ix2 = rb[2], iy2 = rb[3];

        float w, hgt, inter, iou;
        w     = fmaxf(fminf(ix2, jx2L) - fmaxf(ix1, jx1L), 0.f);
        hgt   = fmaxf(fminf(iy2, jy2L) - fmaxf(iy1, jy1L), 0.f);
        inter = w * hgt;
        iou   = inter / (dL[v] - inter + 1e-9f);
        balL[v] = (unsigned)__ballot(iou > IOU_THR);

        w     = fmaxf(fminf(ix2, jx2R) - fmaxf(ix1, jx1R), 0.f);
        hgt   = fmaxf(fminf(iy2, jy2R) - fmaxf(iy1, jy1R), 0.f);
        inter = w * hgt;
        iou   = inter / (dR[v] - inter + 1e-9f);
        balR[v] = (unsigned)__ballot(iou > IOU_THR);
    }

    if (lane == 0) {
        unsigned* mrow = mask + (size_t)b * KPAD * (KPAD / 32);
        #pragma unroll
        for (int v = 0; v < 8; ++v) {
            // ballot bits 0-15: row v, cols 0-15; bits 16-31: row v+8, cols 0-15
            unsigned wlo = (balL[v] & 0xFFFFu) | ((balR[v] & 0xFFFFu) << 16);
            unsigned whi = (balL[v] >> 16)     | ((balR[v] >> 16)     << 16);
            mrow[(size_t)(i0 + v)     * (KPAD / 32) + tj] = wlo;
            mrow[(size_t)(i0 + v + 8) * (KPAD / 32) + tj] = whi;
        }
    }
}

// --------------------------------------------------------------------- K5
__global__ __launch_bounds__(32) void nms_kernel(const unsigned* __restrict__ mask,
                                                 const float* __restrict__ topv,
                                                 const float* __restrict__ boxes_raw,
                                                 const int* __restrict__ classes,
                                                 float* __restrict__ out) {
    int b = blockIdx.x, t = threadIdx.x;   // one wave32 per batch; lane t owns word t
    const unsigned* m = mask + (size_t)b * KPAD * (KPAD / 32);
    const float* tv = topv + (size_t)b * KPAD;

    unsigned validw = 0;
    #pragma unroll 4
    for (int bit = 0; bit < 32; ++bit) {
        int j = t * 32 + bit;
        if (j < KTOP && tv[j] >= CLS_THR) validw |= (1u << bit);
    }

    unsigned rem = 0;
    for (int i = 0; i < KTOP; ++i) {
        int wi = i >> 5, bi = i & 31;
        unsigned remw = __shfl(rem, wi);
        unsigned valw = __shfl(validw, wi);
        bool keep_i = ((valw >> bi) & 1u) && !((remw >> bi) & 1u);
        __builtin_prefetch(&m[(size_t)(i + 1) * (KPAD / 32) + t], 0, 0);
        if (keep_i) {    // wave-uniform branch
            unsigned mw = m[(size_t)i * (KPAD / 32) + t];
            unsigned allowed = (t > wi) ? 0xFFFFFFFFu
                             : ((t == wi && bi < 31) ? (0xFFFFFFFFu << (bi + 1)) : 0u);
            rem |= mw & allowed;           // suppress only j > i
        }
    }
    unsigned keepw = validw & ~rem;

    const float* br = boxes_raw + (size_t)b * KPAD * 4;
    const int*   cl = classes   + (size_t)b * KPAD;
    float* preds    = out;                                   // (B, KTOP, 6)
    float* keep_out = out + (size_t)BATCH * KTOP * 6;        // (B, KTOP)
    for (int bit = 0; bit < 32; ++bit) {
        int j = t * 32 + bit;
        if (j >= KTOP) break;
        float kb = ((keepw >> bit) & 1u) ? 1.f : 0.f;
        size_t po = ((size_t)b * KTOP + j) * 6;
        preds[po + 0] = kb * br[(size_t)j * 4 + 0];
        preds[po + 1] = kb * br[(size_t)j * 4 + 1];
        preds[po + 2] = kb * br[(size_t)j * 4 + 2];
        preds[po + 3] = kb * br[(size_t)j * 4 + 3];
        preds[po + 4] = kb * tv[j];
        preds[po + 5] = kb * (float)cl[j];
        keep_out[(size_t)b * KTOP + j] = kb;
    }
}

// ---------------------------------------------------------------------------
extern "C" void kernel_launch(void* const* d_in, const int* in_sizes, int n_in,
                              void* d_out, int out_size, void* d_ws, size_t ws_size,
                              hipStream_t stream) {
    (void)in_sizes; (void)n_in; (void)out_size;

    DetPtrs P;
    for (int i = 0; i < 5; ++i) {
        P.cls[i] = (const float*)d_in[3 * i + 0];
        P.cnt[i] = (const float*)d_in[3 * i + 1];
        P.reg[i] = (const float*)d_in[3 * i + 2];
    }

    // Workspace carve-up (~8.6 MB total, all 4B aligned)
    float*    scores    = (float*)d_ws;                          // B*NPOS
    float*    topv      = scores + (size_t)BATCH * NPOS;         // B*KPAD
    int*      topi      = (int*)(topv + (size_t)BATCH * KPAD);   // B*KPAD
    int*      classes   = topi + (size_t)BATCH * KPAD;           // B*KPAD
    float*    areasp    = (float*)(classes + (size_t)BATCH * KPAD);
    float*    boxes_raw = areasp + (size_t)BATCH * KPAD;         // B*KPAD*4
    float*    boxes_off = boxes_raw + (size_t)BATCH * KPAD * 4;  // B*KPAD*4
    unsigned* maskp     = (unsigned*)(boxes_off + (size_t)BATCH * KPAD * 4); // B*KPAD*32

    size_t need = ((size_t)BATCH * NPOS + (size_t)BATCH * KPAD * 12
                   + (size_t)BATCH * KPAD * (KPAD / 32)) * 4;
    if (ws_size < need) return;

    zero_kernel<<<(BATCH * KPAD * 4 + 255) / 256, 256, 0, stream>>>(topv, areasp, boxes_off);

    dim3 g1((NPOS + 255) / 256, BATCH);
    score_kernel<<<g1, 256, 0, stream>>>(P, scores);

    topk_kernel<<<BATCH, 256, 0, stream>>>(scores, topv, topi);

    dim3 g3((KTOP + 255) / 256, BATCH);
    gather_kernel<<<g3, 256, 0, stream>>>(P, topi, boxes_raw, boxes_off, areasp, classes);

    dim3 g4(KPAD / 32, (KTOP + 15) / 16, BATCH);   // 32 x 63 x 32 tiles
    ioumask_kernel<<<g4, 32, 0, stream>>>(boxes_off, areasp, maskp);

    nms_kernel<<<BATCH, 32, 0, stream>>>(maskp, topv, boxes_raw, classes, (float*)d_out);
}